// Router_88493506167189
// MI455X (gfx1250) — compile-verified
//
#include <hip/hip_runtime.h>
#include <cstdint>
#include <cstddef>

// MoE router: logits = x @ W^T + b ; top-2 ; sparse softmax ; ids
// x: (4,4096,2048) f32, W: (64,2048) f32, b: (64,) f32
// out: router_output (16384,64) f32  ||  ids (16384,2) i32 (concatenated)

typedef __attribute__((ext_vector_type(2))) float v2f;
typedef __attribute__((ext_vector_type(4))) float v4f;
typedef __attribute__((ext_vector_type(8))) float v8f;

constexpr int Bb    = 4;
constexpr int Ss    = 4096;
constexpr int Dd    = 2048;
constexpr int Ee    = 64;
constexpr int NROWS = Bb * Ss;   // 16384 token rows
constexpr int RPB   = 64;        // token rows per block (16 per wave x 4 waves)
constexpr int KC    = 64;        // K chunk staged in LDS (per buffer)
constexpr int AS    = 68;        // LDS row stride (floats): 68%64=4 -> banks 4m..4m+3
constexpr int LS    = 65;        // logits LDS row stride (floats)
constexpr int BUF   = RPB * AS + Ee * AS;   // floats per double-buffer slot (8704)
constexpr int NCH   = Dd / KC;   // 32 K-chunks

// Async DMA: global -> LDS, 16B per lane, tracked by ASYNCcnt (no staging VGPRs).
__device__ __forceinline__ void async_b128_to_lds(const float* gsrc, const float* ldst) {
    unsigned           l = (unsigned)(uintptr_t)ldst;           // low 32b = LDS byte addr
    unsigned long long g = (unsigned long long)(uintptr_t)gsrc;
    asm volatile("global_load_async_to_lds_b128 %0, %1, off"
                 :: "v"(l), "v"(g) : "memory");
}

__global__ __launch_bounds__(128, 1) void router_wmma_kernel(
    const float* __restrict__ x,
    const float* __restrict__ W,
    const float* __restrict__ bias,
    float* __restrict__ out,
    int* __restrict__ ids)
{
    __shared__ float lds[2 * BUF];   // 69,632 bytes -> 4 blocks/WGP

    const int tid     = threadIdx.x;
    const int wave    = tid >> 5;
    const int lane    = tid & 31;
    const int rowBase = blockIdx.x * RPB;

    // WMMA fragment coordinates (ISA 16x4 f32 A-layout, mirrored for B)
    const int m  = lane & 15;          // M for A-frag, N for B-frag
    const int kb = (lane >> 4) << 1;   // K base: 0 (lanes 0-15) or 2 (lanes 16-31)

    // per-lane bias for the 4 expert tiles (expert n = 16*t + m)
    float bval[4];
    #pragma unroll
    for (int t = 0; t < 4; ++t) bval[t] = bias[t * 16 + m];

    v8f c[4] = {};   // 16 rows x {experts 16t..16t+15}, t = 0..3

    // Issue one chunk's staging: 64x64 x-tile + 64x64 W-tile, 16 asyncs/thread.
    auto stage = [&](float* buf, int kc) {
        float* bufA = buf;
        float* bufW = buf + RPB * AS;
        #pragma unroll
        for (int i = 0; i < 8; ++i) {
            const int f  = tid + i * 128;    // 0..1023
            const int r  = f >> 4;           // row 0..63
            const int kv = (f & 15) << 2;    // float offset 0..60, step 4
            async_b128_to_lds(x + (size_t)(rowBase + r) * Dd + kc + kv,
                              bufA + r * AS + kv);
            async_b128_to_lds(W + (size_t)r * Dd + kc + kv,
                              bufW + r * AS + kv);
        }
    };

    // Software pipeline: prefetch chunk ci+1 while computing chunk ci.
    stage(lds, 0);
    for (int ci = 0; ci < NCH; ++ci) {
        float* cur = lds + (ci & 1) * BUF;
        float* nxt = lds + ((ci + 1) & 1) * BUF;
        if (ci + 1 < NCH) {
            stage(nxt, (ci + 1) * KC);
            // 16 newest outstanding asyncs belong to chunk ci+1; in-order
            // completion => asynccnt<=16 guarantees chunk ci is resident.
            asm volatile("s_wait_asynccnt 16" ::: "memory");
        } else {
            asm volatile("s_wait_asynccnt 0" ::: "memory");
        }
        __syncthreads();   // chunk ci resident for all waves

        const float* aP = cur + (wave * 16 + m) * AS + kb;
        const float* wP = cur + RPB * AS + m * AS + kb;
        #pragma unroll 4
        for (int k = 0; k < KC; k += 4) {
            v2f a = *(const v2f*)(aP + k);
            #pragma unroll
            for (int t = 0; t < 4; ++t) {
                v2f bfrag = *(const v2f*)(wP + t * 16 * AS + k);
                c[t] = __builtin_amdgcn_wmma_f32_16x16x4_f32(
                    /*neg_a=*/false, a, /*neg_b=*/false, bfrag,
                    /*c_mod=*/(short)0, c[t],
                    /*reuse_a=*/false, /*reuse_b=*/false);
            }
        }
        __syncthreads();   // all waves done reading before buffer reuse
    }

    // Dump biased logits: wave-private 16x64 region (stride 65, conflict-free).
    float* lg = lds + wave * 16 * LS;
    const int mh = (lane >> 4) << 3;   // +8 row offset for upper half-lanes
    #pragma unroll
    for (int t = 0; t < 4; ++t) {
        #pragma unroll
        for (int v = 0; v < 8; ++v) {
            lg[(v + mh) * LS + t * 16 + m] = c[t][v] + bval[t];
        }
    }

    // Lanes 0-15: one token row each -> top-2, 2-way softmax, scatter, ids.
    if (lane < 16) {
        const float* rowL = lg + lane * LS;
        float v1 = -3.402823466e38f, v2 = -3.402823466e38f;
        int i1 = 0, i2 = 0;
        #pragma unroll
        for (int j = 0; j < Ee; ++j) {
            float v = rowL[j];
            if (v > v1)      { v2 = v1; i2 = i1; v1 = v; i1 = j; }
            else if (v > v2) { v2 = v;  i2 = j; }
        }
        const float e   = __expf(v2 - v1);     // <= 1, numerically safe
        const float inv = 1.0f / (1.0f + e);
        const int grow  = rowBase + wave * 16 + lane;

        float* orow = out + (size_t)grow * Ee;
        #pragma unroll
        for (int j = 0; j < Ee; j += 4) {
            *(v4f*)(orow + j) = (v4f){0.0f, 0.0f, 0.0f, 0.0f};
        }
        orow[i1] = inv;        // softmax over {v1, v2}; zeros elsewhere
        orow[i2] = e * inv;

        ids[grow * 2 + 0] = i1;
        ids[grow * 2 + 1] = i2;
    }
}

extern "C" void kernel_launch(void* const* d_in, const int* in_sizes, int n_in,
                              void* d_out, int out_size, void* d_ws, size_t ws_size,
                              hipStream_t stream) {
    (void)in_sizes; (void)n_in; (void)d_ws; (void)ws_size; (void)out_size;
    const float* x = (const float*)d_in[0];
    const float* W = (const float*)d_in[1];
    const float* b = (const float*)d_in[2];
    float* out = (float*)d_out;
    int*   ids = (int*)(out + (size_t)NROWS * Ee);   // ids follow router_output

    dim3 grid(NROWS / RPB);   // 256 blocks
    dim3 block(128);          // 4 waves (wave32)
    hipLaunchKernelGGL(router_wmma_kernel, grid, block, 0, stream,
                       x, W, b, out, ids);
}